// Model_42803644072532
// MI455X (gfx1250) — compile-verified
//
#include <hip/hip_runtime.h>
#include <hip/hip_bf16.h>

// ---------------- constants (match reference) ----------------
#define NN    50000
#define EE    800000
#define F_IN  128
#define HID   64          // C per head
#define HEADS 4
#define OUTD  10
#define GG    64
#define NEG_SLOPE 0.2f

typedef float v2f __attribute__((ext_vector_type(2)));
typedef float v8f __attribute__((ext_vector_type(8)));
typedef unsigned int u32v4 __attribute__((ext_vector_type(4)));
typedef int i32v4 __attribute__((ext_vector_type(4)));
typedef int i32v8 __attribute__((ext_vector_type(8)));

#ifndef __has_builtin
#define __has_builtin(x) 0
#endif
#if __has_builtin(__builtin_amdgcn_tensor_load_to_lds) && \
    __has_builtin(__builtin_amdgcn_s_wait_tensorcnt)
#define USE_TDM 1
#else
#define USE_TDM 0
#endif

static inline int cdiv_host(long a, long b) { return (int)((a + b - 1) / b); }

// ---------------- generic fill ----------------
__global__ void fill_f32(float* __restrict__ p, float v, int n) {
    int t = blockIdx.x * blockDim.x + threadIdx.x;
    if (t < n) p[t] = v;
}

// =====================================================================
// f32 WMMA GEMM with TDM-staged weight panels.
//   C[M,Nc] = A[M,K] @ B[K,Nc]
// Block = 160 threads = 5 waves; each wave owns one 16-row M-tile, all five
// share one 16-col N-tile whose B panel is staged through LDS in KC-row
// chunks (double buffered, fetched by the Tensor Data Mover).
// WMMA fragment layouts per CDNA5 ISA:
//   A frag (16x4): lane l: M=l%16, VGPR j holds K = j + 2*(l/16)
//   B frag (4x16): lane l: N=l%16, VGPR j holds K = j + 2*(l/16)
//   C/D (16x16):   lane l: N=l%16, VGPR j holds M = j + 8*(l/16)
// =====================================================================
#define KC     64
#define GWAVES 5

__device__ __forceinline__ void stage_chunk(const float* __restrict__ B,
                                            int K, int Nc, int tn, int k0,
                                            float* dst) {
#if USE_TDM
    if (threadIdx.x == 0) {
        unsigned long long g =
            (unsigned long long)(const void*)(B + (size_t)k0 * Nc + tn * 16);
        unsigned lds = (unsigned)(uintptr_t)dst;  // low 32 bits = LDS offset
        // D# group 0: count=1 | lds_addr | global_addr[56:0] | type=2
        u32v4 g0;
        g0.x = 1u;
        g0.y = lds;
        g0.z = (unsigned)g;
        g0.w = (unsigned)((g >> 32) & 0x01FFFFFFull) | 0x80000000u; // type=2
        // D# group 1: data_size=4B, tensor_dim0=Nc, tensor_dim1=K,
        //             tile_dim0=16, tile_dim1=KC, tensor_dim0_stride=Nc
        i32v8 g1;
        g1[0] = (int)(2u << 16);                                   // data_size
        g1[1] = (int)(((unsigned)Nc & 0xFFFFu) << 16);             // dim0 lo16
        g1[2] = (int)(((unsigned)Nc >> 16) | (((unsigned)K & 0xFFFFu) << 16));
        g1[3] = (int)(((unsigned)K >> 16) | (16u << 16));          // tile_dim0
        g1[4] = KC;                                                // tile_dim1
        g1[5] = Nc;                                                // dim0 stride
        g1[6] = 0;
        g1[7] = 0;
        i32v4 z4 = {0, 0, 0, 0};
#if __clang_major__ >= 23
        i32v8 z8 = {};
        __builtin_amdgcn_tensor_load_to_lds(g0, g1, z4, z4, z8, 0);
#else
        __builtin_amdgcn_tensor_load_to_lds(g0, g1, z4, z4, 0);
#endif
    }
#else
    // cooperative fallback: plain loads into LDS
    for (int i = threadIdx.x; i < KC * 8; i += blockDim.x) {
        int r = i >> 3, p = (i & 7) * 2;
        *(v2f*)(dst + r * 16 + p) =
            *(const v2f*)(B + (size_t)(k0 + r) * Nc + tn * 16 + p);
    }
#endif
}

__global__ void gemm_wmma_f32(const float* __restrict__ A,
                              const float* __restrict__ B,
                              float* __restrict__ C,
                              int M, int K, int Nc) {
    __shared__ float Bs[2][KC * 16];     // 8 KB double-buffered weight panel

    int tilesN = Nc >> 4;
    int mb = blockIdx.x / tilesN;
    int tn = blockIdx.x - mb * tilesN;
    int wv   = threadIdx.x >> 5;
    int lane = threadIdx.x & 31;
    int tm   = mb * GWAVES + wv;         // M % (16*GWAVES) == 0 -> always valid
    int half = lane >> 4;                // 0 or 1
    int l16  = lane & 15;

    const float* Arow = A + (size_t)(tm * 16 + l16) * K;

    stage_chunk(B, K, Nc, tn, 0, &Bs[0][0]);

    v8f acc = {};
    int nch = K / KC;
    for (int c = 0; c < nch; ++c) {
#if USE_TDM
        __builtin_amdgcn_s_wait_tensorcnt(0);   // TDM chunk c landed (wave 0)
#endif
        __syncthreads();                        // publish LDS to all waves
        if (c + 1 < nch)                        // prefetch next chunk
            stage_chunk(B, K, Nc, tn, (c + 1) * KC, &Bs[(c + 1) & 1][0]);

        const float* Bb = &Bs[c & 1][0];
        int kb = c * KC;
#pragma unroll 4
        for (int kk = 0; kk < KC; kk += 4) {
            v2f a = *(const v2f*)(Arow + kb + kk + 2 * half);
            v2f b;
            b.x = Bb[(kk + 2 * half) * 16 + l16];
            b.y = Bb[(kk + 1 + 2 * half) * 16 + l16];
            acc = __builtin_amdgcn_wmma_f32_16x16x4_f32(
                      false, a, false, b, (short)0, acc, false, false);
        }
        __syncthreads();                        // done with Bs[c&1] before reuse
    }

    float* Cp = C + (tn * 16 + l16);
#pragma unroll
    for (int j = 0; j < 8; ++j) {
        Cp[(size_t)(tm * 16 + j + 8 * half) * Nc] = acc[j];
    }
}

// ---------------- attention coefficients: a_src/a_dst [N,H] ----------------
__global__ void attn_coef(const float* __restrict__ h,
                          const float* __restrict__ att_src,
                          const float* __restrict__ att_dst,
                          float* __restrict__ as_out,
                          float* __restrict__ ad_out,
                          int N, int H, int C) {
    int t = blockIdx.x * blockDim.x + threadIdx.x;   // n*H + head
    if (t >= N * H) return;
    int n  = t / H;
    int hh = t - n * H;
    const float* hp = h + (size_t)(n * H + hh) * C;
    const float* as = att_src + hh * C;
    const float* ad = att_dst + hh * C;
    float s = 0.f, d = 0.f;
    for (int c = 0; c < C; ++c) {
        float v = hp[c];
        s += v * as[c];
        d += v * ad[c];
    }
    as_out[t] = s;
    ad_out[t] = d;
}

__device__ __forceinline__ float leaky(float v) {
    return v > 0.f ? v : NEG_SLOPE * v;
}

__device__ __forceinline__ void atomicMaxFloat(float* addr, float value) {
    if (value >= 0.f)
        atomicMax((int*)addr, __float_as_int(value));
    else
        atomicMin((unsigned int*)addr, __float_as_uint(value));
}

// ---------------- pass 1: per-dst per-head running max ----------------
__global__ void edge_max(const int* __restrict__ ei,
                         const float* __restrict__ as,
                         const float* __restrict__ ad,
                         float* __restrict__ amax,
                         int E, int N, int H) {
    int e = blockIdx.x * blockDim.x + threadIdx.x;
    if (e >= E + N) return;
    int s = (e < E) ? ei[e]     : (e - E);   // self-loop tail
    int d = (e < E) ? ei[E + e] : (e - E);
    for (int h = 0; h < H; ++h) {
        float v = leaky(as[s * H + h] + ad[d * H + h]);
        atomicMaxFloat(&amax[d * H + h], v);
    }
}

// ---------------- pass 2: exp(alpha - max), cache it, accumulate denom ----------------
__global__ void edge_exp(const int* __restrict__ ei,
                         const float* __restrict__ as,
                         const float* __restrict__ ad,
                         const float* __restrict__ amax,
                         float* __restrict__ alphaE,
                         float* __restrict__ denom,
                         int E, int N, int H) {
    int e = blockIdx.x * blockDim.x + threadIdx.x;
    if (e >= E + N) return;
    int s = (e < E) ? ei[e]     : (e - E);
    int d = (e < E) ? ei[E + e] : (e - E);
    for (int h = 0; h < H; ++h) {
        float v = leaky(as[s * H + h] + ad[d * H + h]);
        float a = __expf(v - amax[d * H + h]);
        alphaE[e * H + h] = a;
        atomicAdd(&denom[d * H + h], a);
    }
}

// ---------------- pass 3: weighted scatter-add of messages ----------------
__global__ void edge_agg(const int* __restrict__ ei,
                         const float* __restrict__ hsrc,
                         const float* __restrict__ alphaE,
                         const float* __restrict__ denom,
                         float* __restrict__ out,
                         int E, int N, int H, int C) {
    int t = blockIdx.x * blockDim.x + threadIdx.x;
    int c = t & (C - 1);      // C == 64
    int e = t >> 6;
    if (e >= E + N) return;
    int s = (e < E) ? ei[e]     : (e - E);
    int d = (e < E) ? ei[E + e] : (e - E);
    for (int h = 0; h < H; ++h) {
        float w = alphaE[e * H + h] / (denom[d * H + h] + 1e-16f);
        float m = hsrc[(size_t)(s * H + h) * C + c] * w;
        atomicAdd(&out[(size_t)(d * H + h) * C + c], m);
    }
}

// ---------------- bias + relu in place ----------------
__global__ void bias_relu(float* __restrict__ p, const float* __restrict__ b,
                          int N, int F) {
    int t = blockIdx.x * blockDim.x + threadIdx.x;
    if (t >= N * F) return;
    float v = p[t] + b[t % F];
    p[t] = v > 0.f ? v : 0.f;
}

// ---------------- segment-sum pooling via LDS accumulation ----------------
#define NODES_PER_BLOCK 1024
__global__ void pool_sum(const float* __restrict__ h3,
                         const int* __restrict__ batch,
                         float* __restrict__ pooled,
                         int N, int C, int G) {
    __shared__ float lp[GG * HID];   // 16KB
    for (int i = threadIdx.x; i < G * C; i += blockDim.x) lp[i] = 0.f;
    __syncthreads();
    int n0 = blockIdx.x * NODES_PER_BLOCK;
    int total = NODES_PER_BLOCK * C;
    for (int i = threadIdx.x; i < total; i += blockDim.x) {
        int n = n0 + (i >> 6);       // C == 64
        int c = i & (C - 1);
        if (n < N) {
            atomicAdd(&lp[batch[n] * C + c], h3[(size_t)n * C + c]);
        }
    }
    __syncthreads();
    for (int i = threadIdx.x; i < G * C; i += blockDim.x) {
        float v = lp[i];
        if (v != 0.f) atomicAdd(&pooled[i], v);
    }
}

// ---------------- tiny MLP head ----------------
__global__ void mlp_head(const float* __restrict__ pooled,
                         const float* __restrict__ Wm1, const float* __restrict__ bm1,
                         const float* __restrict__ Wm2, const float* __restrict__ bm2,
                         float* __restrict__ out, int G, int C, int O) {
    int g = threadIdx.x;
    if (g >= G) return;
    float hid[HID];
    for (int j = 0; j < C; ++j) {
        float acc = bm1[j];
        for (int k = 0; k < C; ++k) acc += pooled[g * C + k] * Wm1[k * C + j];
        hid[j] = acc > 0.f ? acc : 0.f;
    }
    for (int o = 0; o < O; ++o) {
        float acc = bm2[o];
        for (int k = 0; k < C; ++k) acc += hid[k] * Wm2[k * O + o];
        out[g * O + o] = acc;
    }
}

// =====================================================================
extern "C" void kernel_launch(void* const* d_in, const int* in_sizes, int n_in,
                              void* d_out, int out_size, void* d_ws, size_t ws_size,
                              hipStream_t stream) {
    const float* x    = (const float*)d_in[0];
    const int*   ei   = (const int*)  d_in[1];   // [2,E] flat
    const int*   batch= (const int*)  d_in[2];
    const float* W1 = (const float*)d_in[3];
    const float* b1 = (const float*)d_in[4];
    const float* as1= (const float*)d_in[5];
    const float* ad1= (const float*)d_in[6];
    const float* W2 = (const float*)d_in[7];
    const float* b2 = (const float*)d_in[8];
    const float* as2= (const float*)d_in[9];
    const float* ad2= (const float*)d_in[10];
    const float* W3 = (const float*)d_in[11];
    const float* b3 = (const float*)d_in[12];
    const float* as3= (const float*)d_in[13];
    const float* ad3= (const float*)d_in[14];
    const float* Wm1= (const float*)d_in[15];
    const float* bm1= (const float*)d_in[16];
    const float* Wm2= (const float*)d_in[17];
    const float* bm2= (const float*)d_in[18];
    float* out = (float*)d_out;

    // ---- carve workspace ----
    char*  base = (char*)d_ws;
    size_t off  = 0;
    auto alloc = [&](size_t bytes) -> float* {
        float* p = (float*)(base + off);
        off = (off + bytes + 255) & ~(size_t)255;
        return p;
    };
    const int Ep = EE + NN;                 // with self-loops
    float* hA     = alloc((size_t)NN * HEADS * HID * sizeof(float)); // gemm out
    float* hB     = alloc((size_t)NN * HEADS * HID * sizeof(float)); // agg out / next in
    float* h3     = alloc((size_t)NN * HID * sizeof(float));         // layer3 out
    float* a_src  = alloc((size_t)NN * HEADS * sizeof(float));
    float* a_dst  = alloc((size_t)NN * HEADS * sizeof(float));
    float* amax   = alloc((size_t)NN * HEADS * sizeof(float));
    float* denom  = alloc((size_t)NN * HEADS * sizeof(float));
    float* alphaE = alloc((size_t)Ep * HEADS * sizeof(float));
    float* pooled = alloc((size_t)GG * HID * sizeof(float));
    (void)ws_size; (void)in_sizes; (void)n_in; (void)out_size;

    const int TB = 256;

    auto gat_layer = [&](const float* hin, int K,
                         const float* W, const float* b,
                         const float* as_w, const float* ad_w,
                         int H, float* gout, float* aout) {
        const int C = HID, F = H * C;
        // GEMM: 5 waves per block (NN = 50000 = 625 * 5 * 16, exact)
        {
            int tilesN  = F / 16;
            int mBlocks = NN / (16 * GWAVES);
            gemm_wmma_f32<<<tilesN * mBlocks, 32 * GWAVES, 0, stream>>>(
                hin, W, gout, NN, K, F);
        }
        attn_coef<<<cdiv_host((long)NN * H, TB), TB, 0, stream>>>(gout, as_w, ad_w,
                                                                  a_src, a_dst, NN, H, C);
        fill_f32<<<cdiv_host((long)NN * H, TB), TB, 0, stream>>>(amax, -3.0e38f, NN * H);
        fill_f32<<<cdiv_host((long)NN * H, TB), TB, 0, stream>>>(denom, 0.f, NN * H);
        fill_f32<<<cdiv_host((long)NN * F, TB), TB, 0, stream>>>(aout, 0.f, NN * F);
        edge_max<<<cdiv_host(Ep, TB), TB, 0, stream>>>(ei, a_src, a_dst, amax, EE, NN, H);
        edge_exp<<<cdiv_host(Ep, TB), TB, 0, stream>>>(ei, a_src, a_dst, amax,
                                                       alphaE, denom, EE, NN, H);
        edge_agg<<<cdiv_host((long)Ep * C, TB), TB, 0, stream>>>(ei, gout, alphaE, denom,
                                                                 aout, EE, NN, H, C);
        bias_relu<<<cdiv_host((long)NN * F, TB), TB, 0, stream>>>(aout, b, NN, F);
    };

    gat_layer(x,  F_IN,        W1, b1, as1, ad1, HEADS, hA, hB);
    gat_layer(hB, HEADS * HID, W2, b2, as2, ad2, HEADS, hA, hB);
    gat_layer(hB, HEADS * HID, W3, b3, as3, ad3, 1,     hA, h3);

    fill_f32<<<cdiv_host((long)GG * HID, TB), TB, 0, stream>>>(pooled, 0.f, GG * HID);
    pool_sum<<<cdiv_host(NN, NODES_PER_BLOCK), TB, 0, stream>>>(h3, batch, pooled,
                                                                NN, HID, GG);
    mlp_head<<<1, GG, 0, stream>>>(pooled, Wm1, bm1, Wm2, bm2, out, GG, HID, OUTD);
}